// SingleGPUMoETorchFFN_42786464203358
// MI455X (gfx1250) — compile-verified
//
#include <hip/hip_runtime.h>
#include <hip/hip_bf16.h>
#include <math.h>

// ---------------- problem constants (from reference setup) ----------------
constexpr int T = 8192;           // B*S tokens
constexpr int D = 2048;           // model dim
constexpr int H = 5632;           // hidden dim
constexpr int E = 8;              // experts
constexpr int TOPK = 2;
constexpr int TILE_M = 16;        // rows per WMMA tile (expert-aligned padding)
constexpr int MAX_ROWS = T * TOPK + E * TILE_M;   // 16512 compact rows worst case
constexpr int MAX_MTILES = MAX_ROWS / TILE_M;     // 1032
constexpr int KCHUNK = 512;       // LDS staging chunk (32 KB fp32 for 16 rows)

typedef __attribute__((ext_vector_type(2))) float v2f;
typedef __attribute__((ext_vector_type(8))) float v8f;
typedef __attribute__((ext_vector_type(4))) unsigned int v4u;
typedef __attribute__((ext_vector_type(8))) unsigned int v8u;

// D(16x16,f32) = A(16x4,f32) x B(4x16,f32) + C   — CDNA5 fp32 WMMA
__device__ __forceinline__ v8f wmma4(v2f a, v2f b, v8f c) {
  return __builtin_amdgcn_wmma_f32_16x16x4_f32(
      /*neg_a=*/false, a, /*neg_b=*/false, b,
      /*c_mod=*/(short)0, c, /*reuse_a=*/false, /*reuse_b=*/false);
}

// Tensor Data Mover: DMA a tile described by D# (4 SGPR groups) into LDS.
// Tracked by TENSORcnt. EXEC is ignored; issue from one wave per block.
__device__ __forceinline__ void tdm_load(v4u g0, v8u g1, v4u g2, v4u g3) {
  asm volatile("tensor_load_to_lds %0, %1, %2, %3"
               :: "s"(g0), "s"(g1), "s"(g2), "s"(g3)
               : "memory");
}
__device__ __forceinline__ void tdm_wait() {
  __builtin_amdgcn_s_wait_tensorcnt(0);
}

// D# group0: count=1 | optional gather bits, lds_addr, 57b global addr, type=2
__device__ __forceinline__ v4u tdm_g0(unsigned ctrl, unsigned lds_addr,
                                      const void* gptr) {
  unsigned long long ga = (unsigned long long)(size_t)gptr;
  v4u g0 = { ctrl, lds_addr,
             (unsigned)(ga & 0xFFFFFFFFu),
             (unsigned)((ga >> 32) & 0x01FFFFFFu) | 0x80000000u /*type=2*/ };
  return g0;
}

// D# group1 for a 2D fp32 tile: tile_dim0 x tile_dim1, row stride stride0
__device__ __forceinline__ v8u tdm_g1(unsigned tdim0, unsigned tdim1,
                                      unsigned tile0, unsigned tile1,
                                      unsigned stride0) {
  v8u g1 = { 0x00020000u,                           // wg_mask=0, data_size=4B
             (tdim0 & 0xFFFFu) << 16,               // tensor_dim0 lo
             (tdim0 >> 16) | ((tdim1 & 0xFFFFu) << 16),
             (tdim1 >> 16) | (tile0 << 16),         // tile_dim0
             tile1,                                 // tile_dim1 (tile_dim2=0)
             stride0,                               // dim0 stride lo32
             0u, 0u };
  return g1;
}

// ---------------- kernel 0: zero routing counters ----------------
__global__ void moe_init_kernel(int* __restrict__ counts, int* __restrict__ fill) {
  if (threadIdx.x < E) { counts[threadIdx.x] = 0; fill[threadIdx.x] = 0; }
}

// ---------------- kernel 1: zero output (atomically accumulated later) ----
__global__ __launch_bounds__(256) void moe_zero_kernel(float4* __restrict__ y, int n4) {
  for (int i = blockIdx.x * blockDim.x + threadIdx.x; i < n4; i += gridDim.x * blockDim.x)
    y[i] = make_float4(0.f, 0.f, 0.f, 0.f);
}

// ---------------- kernel 2: gating (scores -> top2 -> softmax) ------------
__global__ __launch_bounds__(128) void moe_gate_kernel(
    const float* __restrict__ x, const float* __restrict__ gw,
    int* __restrict__ counts, int* __restrict__ top_idx, float* __restrict__ top_w) {
  const int lane = threadIdx.x & 31;
  const int wid  = threadIdx.x >> 5;
  const int t = blockIdx.x * 4 + wid;           // one wave32 per token
  if (t >= T) return;

  float acc[E];
#pragma unroll
  for (int e = 0; e < E; ++e) acc[e] = 0.f;

  const float2* xp = (const float2*)(x + (size_t)t * D);
  for (int i = lane; i < D / 2; i += 32) {
    float2 xv = xp[i];
#pragma unroll
    for (int e = 0; e < E; ++e) {
      float2 gv = ((const float2*)(gw + (size_t)e * D))[i];
      acc[e] += xv.x * gv.x + xv.y * gv.y;
    }
  }
#pragma unroll
  for (int e = 0; e < E; ++e)
#pragma unroll
    for (int off = 16; off > 0; off >>= 1)
      acc[e] += __shfl_xor(acc[e], off, 32);

  if (lane == 0) {
    int i1 = 0; float s1 = acc[0];
    for (int e = 1; e < E; ++e) if (acc[e] > s1) { s1 = acc[e]; i1 = e; }
    int i2 = -1; float s2 = -INFINITY;
    for (int e = 0; e < E; ++e) if (e != i1 && acc[e] > s2) { s2 = acc[e]; i2 = e; }
    float e2 = __expf(s2 - s1);                 // softmax over the 2 logits
    float inv = 1.f / (1.f + e2);
    top_idx[t * 2 + 0] = i1; top_idx[t * 2 + 1] = i2;
    top_w[t * 2 + 0] = inv;  top_w[t * 2 + 1] = e2 * inv;
    atomicAdd(&counts[i1], 1);
    atomicAdd(&counts[i2], 1);
  }
}

// ---------------- kernel 3: build compact, expert-padded row lists --------
__global__ __launch_bounds__(256) void moe_assign_kernel(
    const int* __restrict__ top_idx, const float* __restrict__ top_w,
    const int* __restrict__ counts, int* __restrict__ fill,
    int* __restrict__ ptotal, int* __restrict__ row_token,
    float* __restrict__ row_w, int* __restrict__ row_expert) {
  __shared__ int sBase[E], sCnt[E];
  if (threadIdx.x == 0) {
    int o = 0;
    for (int e = 0; e < E; ++e) {
      sBase[e] = o;
      sCnt[e] = counts[e];
      o += (sCnt[e] + TILE_M - 1) & ~(TILE_M - 1);   // pad to 16-row tiles
    }
    *ptotal = o;
  }
  __syncthreads();
  for (int a = threadIdx.x; a < T * TOPK; a += blockDim.x) {
    int e = top_idx[a];
    int pos = sBase[e] + atomicAdd(&fill[e], 1);
    row_token[pos]  = a >> 1;
    row_w[pos]      = top_w[a];
    row_expert[pos] = e;
  }
  __syncthreads();
  for (int i = threadIdx.x; i < E * TILE_M; i += blockDim.x) {
    int e = i / TILE_M, j = i % TILE_M;
    int pos  = sBase[e] + sCnt[e] + j;
    int pend = sBase[e] + ((sCnt[e] + TILE_M - 1) & ~(TILE_M - 1));
    if (pos < pend) { row_token[pos] = -1; row_w[pos] = 0.f; row_expert[pos] = e; }
  }
}

// ---------------- kernel 4: h = w * silu(x@w1^T) * (x@w3^T) ---------------
// grid = (MAX_MTILES, H/64); block = 128 (4 waves, one 16x16 tile each).
// X tile staged by TDM *gather mode* (row indices = token ids), double-buffered.
__global__ __launch_bounds__(128) void moe_ffn1_kernel(
    const float* __restrict__ x, const float* __restrict__ w1,
    const float* __restrict__ w3, const int* __restrict__ row_token,
    const float* __restrict__ row_w, const int* __restrict__ row_expert,
    const int* __restrict__ ptotal, float* __restrict__ h) {
  __shared__ __align__(16) float ldsX[2][TILE_M * KCHUNK];   // 2 x 32 KB
  const int r0 = blockIdx.x * TILE_M;
  if (r0 >= *ptotal) return;
  const int e = row_expert[r0];

  const int lane = threadIdx.x & 31;
  const int wid  = threadIdx.x >> 5;
  const int half = lane >> 4;                 // K-pair select
  const int lrow = lane & 15;                 // row (A) / col (B,C)
  const int hbase = blockIdx.y * 64 + wid * 16;

  const float* w1r = w1 + ((size_t)e * H + hbase + lrow) * D;
  const float* w3r = w3 + ((size_t)e * H + hbase + lrow) * D;

  // gather indices: 16 x 16-bit token row ids (pads -> 0; zeroed by row_w=0)
  unsigned pk[8];
#pragma unroll
  for (int j = 0; j < 8; ++j) {
    int t0 = row_token[r0 + 2 * j];
    int t1 = row_token[r0 + 2 * j + 1];
    if (t0 < 0) t0 = 0;
    if (t1 < 0) t1 = 0;
    pk[j] = (unsigned)__builtin_amdgcn_readfirstlane((unsigned)t0 |
                                                     ((unsigned)t1 << 16));
  }
  const v4u g2 = { pk[0], pk[1], pk[2], pk[3] };
  const v4u g3 = { pk[4], pk[5], pk[6], pk[7] };
  // tensor: rows of KCHUNK fp32, row stride D, T rows; tile = 512x16 gather
  const v8u g1 = tdm_g1(KCHUNK, T, KCHUNK, TILE_M, D);
  const unsigned GATHER = 0x80000001u;        // gather_mode=1, idx16, count=1
  const unsigned lbase0 = (unsigned)(size_t)&ldsX[0][0];
  const unsigned lbase1 = (unsigned)(size_t)&ldsX[1][0];

  if (wid == 0) {
    tdm_load(tdm_g0(GATHER, lbase0, x), g1, g2, g3);
    tdm_wait();
  }
  __syncthreads();

  v8f acc1 = {}; v8f acc3 = {};
  int buf = 0;
  for (int kc = 0; kc < D; kc += KCHUNK) {
    if (wid == 0 && kc + KCHUNK < D)          // prefetch next chunk via TDM
      tdm_load(tdm_g0(GATHER, buf ? lbase0 : lbase1, x + kc + KCHUNK),
               g1, g2, g3);
    const float* Xb = &ldsX[buf][0];
#pragma unroll 8
    for (int k = 0; k < KCHUNK; k += 4) {
      v2f a  = *(const v2f*)&Xb[lrow * KCHUNK + k + 2 * half];  // A[m, k..k+1]
      v2f b1 = *(const v2f*)&w1r[kc + k + 2 * half];            // B[k..k+1, n]
      v2f b3 = *(const v2f*)&w3r[kc + k + 2 * half];
      acc1 = wmma4(a, b1, acc1);
      acc3 = wmma4(a, b3, acc3);
    }
    if (wid == 0) tdm_wait();
    __syncthreads();
    buf ^= 1;
  }
#pragma unroll
  for (int v = 0; v < 8; ++v) {
    int m = v + 8 * half;
    float s1 = acc1[v], s3 = acc3[v];
    float hv = (s1 / (1.f + __expf(-s1))) * s3;   // silu(s1) * s3
    hv *= row_w[r0 + m];                          // routing weight fused here
    h[(size_t)(r0 + m) * H + hbase + lrow] = hv;
  }
}

// ---------------- kernel 5: y[token] += h @ w2^T --------------------------
// grid = (MAX_MTILES, D/64); block = 128 (4 waves). h tile staged by TDM
// (plain 2D strided tile), double-buffered.
__global__ __launch_bounds__(128) void moe_ffn2_kernel(
    const float* __restrict__ w2, const float* __restrict__ h,
    const int* __restrict__ row_token, const int* __restrict__ row_expert,
    const int* __restrict__ ptotal, float* __restrict__ y) {
  __shared__ __align__(16) float ldsH[2][TILE_M * KCHUNK];   // 2 x 32 KB
  const int r0 = blockIdx.x * TILE_M;
  if (r0 >= *ptotal) return;
  const int e = row_expert[r0];

  const int lane = threadIdx.x & 31;
  const int wid  = threadIdx.x >> 5;
  const int half = lane >> 4;
  const int lrow = lane & 15;
  const int dbase = blockIdx.y * 64 + wid * 16;

  const float* w2r = w2 + ((size_t)e * D + dbase + lrow) * H;
  const float* hrows = h + (size_t)r0 * H;

  const v8u g1 = tdm_g1(KCHUNK, TILE_M, KCHUNK, TILE_M, H);
  const v4u gz = { 0u, 0u, 0u, 0u };
  const unsigned lbase0 = (unsigned)(size_t)&ldsH[0][0];
  const unsigned lbase1 = (unsigned)(size_t)&ldsH[1][0];

  if (wid == 0) {
    tdm_load(tdm_g0(1u, lbase0, hrows), g1, gz, gz);
    tdm_wait();
  }
  __syncthreads();

  v8f acc = {};
  int buf = 0;
  for (int kc = 0; kc < H; kc += KCHUNK) {                // 5632 = 11 * 512
    if (wid == 0 && kc + KCHUNK < H)
      tdm_load(tdm_g0(1u, buf ? lbase0 : lbase1, hrows + kc + KCHUNK),
               g1, gz, gz);
    const float* Hb = &ldsH[buf][0];
#pragma unroll 8
    for (int k = 0; k < KCHUNK; k += 4) {
      v2f a = *(const v2f*)&Hb[lrow * KCHUNK + k + 2 * half];
      v2f b = *(const v2f*)&w2r[kc + k + 2 * half];
      acc = wmma4(a, b, acc);
    }
    if (wid == 0) tdm_wait();
    __syncthreads();
    buf ^= 1;
  }
#pragma unroll
  for (int v = 0; v < 8; ++v) {
    int m = v + 8 * half;
    int tok = row_token[r0 + m];
    if (tok >= 0) atomicAdd(&y[(size_t)tok * D + dbase + lrow], acc[v]);
  }
}

// ---------------- host-side launch ----------------------------------------
extern "C" void kernel_launch(void* const* d_in, const int* in_sizes, int n_in,
                              void* d_out, int out_size, void* d_ws, size_t ws_size,
                              hipStream_t stream) {
  const float* x  = (const float*)d_in[0];   // [T, D]
  const float* gw = (const float*)d_in[1];   // [E, D]
  const float* w1 = (const float*)d_in[2];   // [E, H, D]
  const float* w2 = (const float*)d_in[3];   // [E, D, H]
  const float* w3 = (const float*)d_in[4];   // [E, H, D]
  float* y = (float*)d_out;                  // [T, D]

  // workspace carve-up (256B aligned chunks)
  char* p = (char*)d_ws;
  auto take = [&](size_t bytes) { char* r = p; p += (bytes + 255) & ~size_t(255); return r; };
  int*   counts     = (int*)  take(E * sizeof(int));
  int*   fill       = (int*)  take(E * sizeof(int));
  int*   ptotal     = (int*)  take(sizeof(int));
  int*   top_idx    = (int*)  take((size_t)T * TOPK * sizeof(int));
  float* top_w      = (float*)take((size_t)T * TOPK * sizeof(float));
  int*   row_token  = (int*)  take((size_t)MAX_ROWS * sizeof(int));
  float* row_w      = (float*)take((size_t)MAX_ROWS * sizeof(float));
  int*   row_expert = (int*)  take((size_t)MAX_ROWS * sizeof(int));
  float* hbuf       = (float*)take((size_t)MAX_ROWS * H * sizeof(float)); // ~372 MB

  moe_init_kernel<<<1, 32, 0, stream>>>(counts, fill);
  moe_zero_kernel<<<4096, 256, 0, stream>>>((float4*)y, T * D / 4);
  moe_gate_kernel<<<T / 4, 128, 0, stream>>>(x, gw, counts, top_idx, top_w);
  moe_assign_kernel<<<1, 256, 0, stream>>>(top_idx, top_w, counts, fill, ptotal,
                                           row_token, row_w, row_expert);
  moe_ffn1_kernel<<<dim3(MAX_MTILES, H / 64), 128, 0, stream>>>(
      x, w1, w3, row_token, row_w, row_expert, ptotal, hbuf);
  moe_ffn2_kernel<<<dim3(MAX_MTILES, D / 64), 128, 0, stream>>>(
      w2, hbuf, row_token, row_expert, ptotal, y);
}